// SSIMLoss_16020228014922
// MI455X (gfx1250) — compile-verified
//
#include <hip/hip_runtime.h>
#include <hip/hip_bf16.h>

// ---------------------------------------------------------------------------
// Fused 3D-SSIM loss for MI455X (gfx1250, wave32, WMMA).
//
// One workgroup = one 16^3 interior tile. It loads the 26^3 halo region of
// both volumes into LDS once, then for each of the 5 SSIM channels
// (a, b, a*a, b*b, a*b) runs the separable 11-tap Gaussian blur along x, y, z
// entirely in LDS. Each 1D blur is a banded 16x28 * 28x16 matmul executed as
// 7 chained V_WMMA_F32_16X16X4_F32 instructions. All fetches are branch-free
// (clamped LDS index + v_cndmask zeroing) and the channel loop is fully
// unrolled so no scalar branch ladders appear between ds_loads and wmma.
// ---------------------------------------------------------------------------

typedef __attribute__((ext_vector_type(2))) float v2f;
typedef __attribute__((ext_vector_type(8))) float v8f;

#define DIMV     192
#define TILE     16
#define HALO     5
#define SRC      26              // TILE + 2*HALO
#define NTHREADS 256
#define NWAVES   8               // 256 / wave32
#define NTILES   12              // 192 / 16
#define NBATCH   2
#define NBLOCKS  (NBATCH * NTILES * NTILES * NTILES)   // 3456
#define NVOXELS  (2.0f * 192.0f * 192.0f * 192.0f)     // 14,155,776

// LDS layout (floats); total 75856 floats = 303,424 bytes (< 320 KB / WGP)
#define OFF_A   0                          // 26^3 = 17576
#define OFF_B   17576                      // 26^3 = 17576
#define OFF_T1  35152                      // 26*32*16 = 13312 (x-blurred, y-padded)
#define OFF_T2  48464                      // 26*16*16 = 6656  (xy-blurred)
#define OFF_CH  55120                      // 5 * 16^3 = 20480 (final blurs)
#define OFF_RED 75600                      // 256 reduction scratch
#define SMEM_FLOATS 75856
#define SMEM_BYTES  (SMEM_FLOATS * 4)

// Gaussian(sigma=1.5, n=11), normalized; g[u] multiplies in[i+u], u=0..10.
__device__ __constant__ float G11[11] = {
    0.00102838f, 0.00759876f, 0.03600077f, 0.10936069f, 0.21300553f,
    0.26601171f,
    0.21300553f, 0.10936069f, 0.03600077f, 0.00759876f, 0.00102838f};

__device__ __forceinline__ float wcoef(int d) {
    return (d >= 0 && d <= 10) ? G11[d] : 0.0f;
}

__device__ __forceinline__ int imin(int a, int b) { return a < b ? a : b; }

// One 16x16 output tile of a 1D conv: D = sum_c A_c(16x4) * B_c(4x16).
// A-frag layout (32b A 16x4): lane m=l&15; VGPR0 holds K=kb, VGPR1 K=kb+1,
// kb = 2*(l>=16).  B mirrored: VGPR0 = B[kb][n], VGPR1 = B[kb+1][n], n=l&15.
// D (32b 16x16 C/D): row = v + 8*(l>=16), col = l&15.
template <typename FetchB, typename StoreD>
__device__ __forceinline__ void conv16(const v2f afrag[7], int lane,
                                       FetchB fetchB, StoreD storeD) {
    const int kb = ((lane >> 4) << 1);   // 0 or 2
    const int n  = lane & 15;
    v8f acc = {};
#pragma unroll
    for (int c = 0; c < 7; ++c) {
        const int t0 = 4 * c + kb;
        v2f b;
        b.x = fetchB(t0, n);
        b.y = fetchB(t0 + 1, n);
        acc = __builtin_amdgcn_wmma_f32_16x16x4_f32(
            false, afrag[c], false, b, (short)0, acc, false, false);
    }
    const int rbase = ((lane >> 4) << 3);   // 0 or 8
#pragma unroll
    for (int v = 0; v < 8; ++v) storeD(rbase + v, n, acc[v]);
}

__global__ __launch_bounds__(NTHREADS, 1)
void ssim_fused_kernel(const float* __restrict__ xin,
                       const float* __restrict__ yin,
                       const float* __restrict__ zin,
                       const int* __restrict__ modality,
                       float* __restrict__ partial) {
    extern __shared__ float smem[];
    float* sA  = smem + OFF_A;
    float* sB  = smem + OFF_B;
    float* t1  = smem + OFF_T1;
    float* t2  = smem + OFF_T2;
    float* ch  = smem + OFF_CH;
    float* red = smem + OFF_RED;

    const int tid  = threadIdx.x;
    const int wave = tid >> 5;
    const int lane = tid & 31;

    const float* __restrict__ A = xin;
    const float* __restrict__ B = (modality[0] == 0) ? zin : yin;

    const int tz = blockIdx.z % NTILES;
    const int nb = blockIdx.z / NTILES;
    const int x0 = blockIdx.x * TILE;
    const int y0 = blockIdx.y * TILE;
    const int z0 = tz * TILE;

    // ---- Stage 0: global -> LDS halo tiles (zero-padded at volume edges) ----
    for (int i = tid; i < SRC * SRC * SRC; i += NTHREADS) {
        const int lx = i % SRC;
        const int ly = (i / SRC) % SRC;
        const int lz = i / (SRC * SRC);
        const int gx = x0 - HALO + lx;
        const int gy = y0 - HALO + ly;
        const int gz = z0 - HALO + lz;
        float va = 0.0f, vb = 0.0f;
        if (gx >= 0 && gx < DIMV && gy >= 0 && gy < DIMV && gz >= 0 && gz < DIMV) {
            const size_t gidx =
                (((size_t)nb * DIMV + (size_t)gz) * DIMV + (size_t)gy) * DIMV +
                (size_t)gx;
            va = A[gidx];
            vb = B[gidx];
        }
        sA[i] = va;
        sB[i] = vb;
    }

    // ---- Per-lane Gaussian A-fragments (shared by all three axes) ----
    v2f afrag[7];
    {
        const int m  = lane & 15;
        const int kb = ((lane >> 4) << 1);
#pragma unroll
        for (int c = 0; c < 7; ++c) {
            const int t = 4 * c + kb;
            v2f a;
            a.x = wcoef(t - m);
            a.y = wcoef(t + 1 - m);
            afrag[c] = a;
        }
    }
    __syncthreads();

    // ---- Stages 1-3 per channel: blur x, blur y, blur z (all via WMMA) ----
#pragma unroll
    for (int cc = 0; cc < 5; ++cc) {
        // Pass X: 26 z-slices * 2 y-column-tiles = 52 tile problems.
        for (int p = wave; p < 52; p += NWAVES) {
            const int zz = p % SRC;
            const int yt = p / SRC;          // 0 or 1
            conv16(afrag, lane,
                [&](int t, int n) -> float {
                    const int y  = yt * 16 + n;
                    const int yc = imin(y, SRC - 1);
                    const int tc = imin(t, SRC - 1);
                    const int idx = (zz * SRC + yc) * SRC + tc;
                    // Unconditional clamped loads; zero invalid lanes by select.
                    const float va = sA[idx];
                    const float vb = sB[idx];
                    const float pv = (cc == 0) ? va
                                   : (cc == 1) ? vb
                                   : (cc == 2) ? va * va
                                   : (cc == 3) ? vb * vb
                                               : va * vb;
                    return (y < SRC && t < SRC) ? pv : 0.0f;
                },
                [&](int row, int n, float val) {
                    const int y = yt * 16 + n;       // may reach 31: pad region
                    t1[(zz * 32 + y) * TILE + row] = val;
                });
        }
        __syncthreads();

        // Pass Y: 26 z-slices.
        for (int p = wave; p < SRC; p += NWAVES) {
            const int zz = p;
            conv16(afrag, lane,
                [&](int t, int n) -> float {
                    const int tc = imin(t, SRC - 1);
                    const float v = t1[(zz * 32 + tc) * TILE + n];
                    return (t < SRC) ? v : 0.0f;
                },
                [&](int row, int n, float val) {
                    t2[zz * 256 + row * TILE + n] = val;
                });
        }
        __syncthreads();

        // Pass Z: 16 y-rows.
        for (int p = wave; p < TILE; p += NWAVES) {
            const int yy = p;
            conv16(afrag, lane,
                [&](int t, int n) -> float {
                    const int tc = imin(t, SRC - 1);
                    const float v = t2[tc * 256 + yy * TILE + n];
                    return (t < SRC) ? v : 0.0f;
                },
                [&](int row, int n, float val) {
                    ch[cc * 4096 + row * 256 + yy * TILE + n] = val;
                });
        }
        __syncthreads();
    }

    // ---- Stage 4: SSIM map + block reduction ----
    const float C1 = 0.01f * 0.01f;
    const float C2 = 0.03f * 0.03f;
    float sum = 0.0f;
    for (int v = tid; v < 4096; v += NTHREADS) {
        const float mu1 = ch[0 * 4096 + v];
        const float mu2 = ch[1 * 4096 + v];
        const float exx = ch[2 * 4096 + v];
        const float eyy = ch[3 * 4096 + v];
        const float exy = ch[4 * 4096 + v];
        const float mu1s = mu1 * mu1;
        const float mu2s = mu2 * mu2;
        const float s11 = exx - mu1s;
        const float s22 = eyy - mu2s;
        const float s12 = exy - mu1 * mu2;
        const float num = (2.0f * mu1 * mu2 + C1) * (2.0f * s12 + C2);
        const float den = (mu1s + mu2s + C1) * (s11 + s22 + C2);
        sum += num / den;
    }
    red[tid] = sum;
    __syncthreads();
#pragma unroll
    for (int s = NTHREADS / 2; s > 0; s >>= 1) {
        if (tid < s) red[tid] += red[tid + s];
        __syncthreads();
    }
    if (tid == 0) {
        const int bid =
            (blockIdx.z * gridDim.y + blockIdx.y) * gridDim.x + blockIdx.x;
        partial[bid] = red[0];
    }
}

__global__ __launch_bounds__(NTHREADS, 1)
void ssim_finalize_kernel(const float* __restrict__ partial,
                          float* __restrict__ out) {
    __shared__ float red[NTHREADS];
    float sum = 0.0f;
    for (int i = threadIdx.x; i < NBLOCKS; i += NTHREADS) sum += partial[i];
    red[threadIdx.x] = sum;
    __syncthreads();
#pragma unroll
    for (int s = NTHREADS / 2; s > 0; s >>= 1) {
        if (threadIdx.x < s) red[threadIdx.x] += red[threadIdx.x + s];
        __syncthreads();
    }
    if (threadIdx.x == 0) {
        const float mean = red[0] / NVOXELS;
        const float d = 1.0f - mean;
        out[0] = d * d;
    }
}

extern "C" void kernel_launch(void* const* d_in, const int* in_sizes, int n_in,
                              void* d_out, int out_size, void* d_ws, size_t ws_size,
                              hipStream_t stream) {
    const float* x = (const float*)d_in[0];
    const float* y = (const float*)d_in[1];
    const float* z = (const float*)d_in[2];
    const int* modality = (const int*)d_in[3];
    float* out = (float*)d_out;
    float* partial = (float*)d_ws;   // NBLOCKS floats

    // Allow > default dynamic LDS (303,424 B; CDNA5 WGP supports 320 KB/WG).
    (void)hipFuncSetAttribute((const void*)ssim_fused_kernel,
                              hipFuncAttributeMaxDynamicSharedMemorySize,
                              SMEM_BYTES);

    dim3 grid(NTILES, NTILES, NTILES * NBATCH);
    ssim_fused_kernel<<<grid, NTHREADS, SMEM_BYTES, stream>>>(
        x, y, z, modality, partial);
    ssim_finalize_kernel<<<1, NTHREADS, 0, stream>>>(partial, out);
}